// PolylineEncoder_34737695490219
// MI455X (gfx1250) — compile-verified
//
#include <hip/hip_runtime.h>
#include <hip/hip_bf16.h>

// ---------------------------------------------------------------------------
// PolylineEncoder for MI455X (gfx1250, wave32)
//   h    = relu(poly[B,P,N,9] @ W1[9,128] + b1)          (VALU, K=9)
//   feat = h @ W2[128,128] + b2                          (v_wmma_f32_16x16x4_f32)
//   out  = max over N of where(mask, feat, -1e9)         (fused into fragments)
//
// One wave handles one 16-point half-polyline; h is produced directly in the
// WMMA A-fragment register layout so layer1->layer2 never touches memory.
// W1, b1, b2 AND W2 columns 0-31 only (16 KB) are staged into the 320 KB WGP LDS via
// the CDNA5 async global->LDS DMA path; B-fragments then come from LDS with
// a 144-float row stride (144 % 64 == 16) so the two half-wave row groups
// hit disjoint bank sets.
// ---------------------------------------------------------------------------

#define BATCH   16
#define PLINES  512
#define NPTS    32
#define CIN     9
#define HID     128
#define W2S     144            /* padded LDS row stride for W2 (bank-conflict-free) */
#define NEG_INF_VAL (-1000000000.0f)

typedef __attribute__((ext_vector_type(2))) float v2f;
typedef __attribute__((ext_vector_type(8))) float v8f;

// LDS byte offset of a __shared__ object (generic -> AS3 -> 32-bit offset)
__device__ __forceinline__ unsigned lds_off32(const void* p) {
  return (unsigned)(unsigned long long)(__attribute__((address_space(3))) const char*)p;
}

__global__ __launch_bounds__(256)
void PolylineEncoder_34737695490219_kernel(
    const float* __restrict__ poly,          // [B,P,N,9]
    const unsigned char* __restrict__ mask,  // [B,P,N] (jax bool = 1 byte)
    const float* __restrict__ W1,            // [9,128]
    const float* __restrict__ b1,            // [128]
    const float* __restrict__ W2,            // [128,128]
    const float* __restrict__ b2,            // [128]
    float* __restrict__ out)                 // [B,P,128]
{
  __shared__ float sW2[HID * W2S];   // 73,728 B  (row stride 144 floats)
  __shared__ float sW1[CIN * HID];   //  4,608 B
  __shared__ float sB1[HID];
  __shared__ float sB2[HID];
  __shared__ float sPart[8][HID];    // per-wave partial column maxima

  const int tid  = threadIdx.x;
  const int wave = tid >> 5;
  const int lane = tid & 31;
  const int hi   = lane >> 4;        // half-wave select (WMMA fragment halves)
  const int lr   = lane & 15;

  // ---- Stage W2 columns 0-31 (16 KB) via async global->LDS DMA, 16 B per transfer ------
  // 1024 16-byte chunks (128 rows x 8), 256 threads -> 4 chunks each.
  #pragma unroll
  for (int it = 0; it < 4; ++it) {
    const int idx = tid + it * 256;          // chunk id
    const int row = idx >> 3;
    const int c4  = (idx & 7) * 4;           // first float of the chunk
    asm volatile("global_load_async_to_lds_b128 %0, %1, off"
                 :: "v"(lds_off32(&sW2[row * W2S + c4])),
                    "v"(W2 + row * HID + c4) : "memory");
  }
  // ---- Stage W1 via async global->LDS DMA (ASYNCcnt-tracked) --------------
  for (int i = tid; i < CIN * HID; i += 256) {
    asm volatile("global_load_async_to_lds_b32 %0, %1, off"
                 :: "v"(lds_off32(&sW1[i])), "v"(W1 + i) : "memory");
  }
  if (tid < HID) { sB1[tid] = b1[tid]; sB2[tid] = b2[tid]; }
  asm volatile("s_wait_asynccnt 0x0" ::: "memory");
  __syncthreads();

  // ---- Which half-polyline does this wave own? ----------------------------
  const int g  = blockIdx.x * 4 + (wave >> 1);  // polyline index in [0, B*P)
  const int mt = wave & 1;                      // M tile: points 0-15 or 16-31
  const int pt = mt * 16 + lr;                  // this lane's point row

  // Point coordinates for row lr (duplicated across the two half-waves)
  const float* xp = poly + ((size_t)g * NPTS + pt) * CIN;
  float x[CIN];
  #pragma unroll
  for (int c = 0; c < CIN; ++c) x[c] = xp[c];

  // Validity bits: bit r (r=0..15) = mask of point mt*16+r in this tile
  const unsigned mb = (unsigned)__ballot(mask[(size_t)g * NPTS + pt] != 0);

  // ---- Layer 1: h = relu(x @ W1 + b1), built directly as A-fragments ------
  // A-frag layout for V_WMMA_F32_16X16X4_F32 (16x4 fp32):
  //   lane l holds row M = l%16; VGPR j of k-step k holds K = 4k + 2*(l/16) + j
  v2f a[32];
  #pragma unroll
  for (int k = 0; k < 32; ++k) {
    const int c0 = 4 * k + 2 * hi;
    float h0 = sB1[c0];
    float h1 = sB1[c0 + 1];
    #pragma unroll
    for (int c = 0; c < CIN; ++c) {
      h0 = fmaf(x[c], sW1[c * HID + c0],     h0);
      h1 = fmaf(x[c], sW1[c * HID + c0 + 1], h1);
    }
    a[k].x = fmaxf(h0, 0.0f);
    a[k].y = fmaxf(h1, 0.0f);
  }

  // ---- Layer 2: feat = h @ W2 + b2 via fp32 WMMA, fused mask + max --------
  #pragma unroll
  for (int nt = 0; nt < 8; ++nt) {
    const int col = nt * 16 + lr;          // output column of this lane

    v8f acc = {0.f, 0.f, 0.f, 0.f, 0.f, 0.f, 0.f, 0.f};
    #pragma unroll
    for (int k = 0; k < 32; ++k) {
      // B-frag (4x16 fp32): VGPR j holds row K = 4k + 2j + (l/16), col = l%16
      v2f bfrag;
      bfrag.x = sW2[(4 * k + 0 + hi) * W2S + col];
      bfrag.y = sW2[(4 * k + 2 + hi) * W2S + col];
      acc = __builtin_amdgcn_wmma_f32_16x16x4_f32(
                false, a[k], false, bfrag, (short)0, acc, false, false);
    }

    // C layout: VGPR v, lane l -> row M = v + 8*(l/16), col = l%16 (+16*nt)
    const float bias = sB2[col];
    float m = NEG_INF_VAL;
    #pragma unroll
    for (int v = 0; v < 8; ++v) {
      const int row = v + 8 * hi;
      const float f = ((mb >> row) & 1u) ? (acc[v] + bias) : NEG_INF_VAL;
      m = fmaxf(m, f);
    }
    // merge the two half-wave row groups (rows 0-7 vs 8-15)
    m = fmaxf(m, __shfl_xor(m, 16, 32));
    if (hi == 0) sPart[wave][nt * 16 + lr] = m;
  }
  __syncthreads();

  // ---- Merge the two M-tile waves of each polyline and store --------------
  if (mt == 0) {
    float* o = out + (size_t)g * HID;
    #pragma unroll
    for (int c = lane; c < HID; c += 32)
      o[c] = fmaxf(sPart[wave][c], sPart[wave + 1][c]);
  }
}

extern "C" void kernel_launch(void* const* d_in, const int* in_sizes, int n_in,
                              void* d_out, int out_size, void* d_ws, size_t ws_size,
                              hipStream_t stream) {
  const float*         poly = (const float*)d_in[0];
  const unsigned char* msk  = (const unsigned char*)d_in[1];
  const float*         W1   = (const float*)d_in[2];
  const float*         b1   = (const float*)d_in[3];
  const float*         W2   = (const float*)d_in[4];
  const float*         b2   = (const float*)d_in[5];
  float*               out  = (float*)d_out;

  const int nPoly  = BATCH * PLINES;       // 8192
  const int blocks = nPoly / 4;            // 4 polylines (8 waves) per block
  hipLaunchKernelGGL(PolylineEncoder_34737695490219_kernel,
                     dim3(blocks), dim3(256), 0, stream,
                     poly, msk, W1, b1, W2, b2, out);
}